// GCN_original_42640435315479
// MI455X (gfx1250) — compile-verified
//
#include <hip/hip_runtime.h>
#include <hip/hip_bf16.h>

typedef __attribute__((ext_vector_type(2))) float v2f;
typedef __attribute__((ext_vector_type(8))) float v8f;

#define HDIM 128
#define NGRAPH_MAX 64

// ---------------- degree / norm ----------------

__global__ void k_deg_init(float* deg, int N) {
    int n = blockIdx.x * blockDim.x + threadIdx.x;
    if (n < N) deg[n] = 2.0f;  // self-loop weight
}

__global__ void k_deg_edges(const int* __restrict__ dst, const float* __restrict__ w,
                            float* deg, int E) {
    int e = blockIdx.x * blockDim.x + threadIdx.x;
    if (e < E) atomicAdd(&deg[dst[e]], w[e]);
}

__global__ void k_deg_finalize(float* deg, int N) {
    int n = blockIdx.x * blockDim.x + threadIdx.x;
    if (n < N) {
        float d = deg[n];
        deg[n] = (d > 0.0f) ? rsqrtf(d) : 0.0f;  // now holds d^-1/2
    }
}

__global__ void k_norm(const int* __restrict__ src, const int* __restrict__ dst,
                       const float* __restrict__ w, const float* __restrict__ dinv,
                       float* __restrict__ norm, int E) {
    int e = blockIdx.x * blockDim.x + threadIdx.x;
    if (e < E) norm[e] = dinv[src[e]] * w[e] * dinv[dst[e]];
}

// ---------------- WMMA GEMM: T = H (N x 128) @ W (128 x 128) ----------------
// One block = one 16-row stripe of H. 8 waves, wave w computes output cols
// [16w, 16w+16). f32 WMMA 16x16x4, accumulate over K=128 in 32 steps.

__global__ __launch_bounds__(256) void k_gemm_wmma(const float* __restrict__ H,
                                                   const float* __restrict__ W,
                                                   float* __restrict__ T, int N) {
    __shared__ float sh[16 * HDIM];
    const int row0 = blockIdx.x * 16;
    const int tid  = threadIdx.x;

    // cooperative load of 16x128 f32 stripe (512 float4s, 2 per thread)
    for (int i = tid; i < 16 * 32; i += 256) {
        int r = i >> 5;    // row in stripe
        int c = i & 31;    // float4 column
        float4 v;
        if (row0 + r < N) v = ((const float4*)(H + (size_t)(row0 + r) * HDIM))[c];
        else              v = make_float4(0.f, 0.f, 0.f, 0.f);
        ((float4*)sh)[i] = v;
    }
    __syncthreads();

    const int wave  = tid >> 5;       // 0..7 -> output column tile
    const int lane  = tid & 31;
    const int col0  = wave * 16;
    const int m     = lane & 15;      // A: row index, B: col index
    const int khalf = (lane >> 4) * 2; // 0 (lanes 0-15) or 2 (lanes 16-31)

    v8f acc = {};
    for (int k0 = 0; k0 < HDIM; k0 += 4) {
        v2f a, b;
        // A 16x4 f32: lane m holds row m; VGPR0/1 = K khalf+0 / khalf+1
        a.x = sh[m * HDIM + k0 + khalf + 0];
        a.y = sh[m * HDIM + k0 + khalf + 1];
        // B 4x16 f32: lane m holds col (col0+m); VGPR0/1 = K khalf+0 / khalf+1
        b.x = W[(k0 + khalf + 0) * HDIM + col0 + m];
        b.y = W[(k0 + khalf + 1) * HDIM + col0 + m];
        acc = __builtin_amdgcn_wmma_f32_16x16x4_f32(
            /*neg_a=*/false, a, /*neg_b=*/false, b,
            /*c_mod=*/(short)0, acc, /*reuse_a=*/false, /*reuse_b=*/false);
    }

    // C/D layout: VGPR i -> M = i + 8*(lane>=16), N = lane&15
    const int mbase = (lane >> 4) * 8;
    #pragma unroll
    for (int i = 0; i < 8; ++i) {
        int r = row0 + mbase + i;
        if (r < N) T[(size_t)r * HDIM + col0 + m] = acc[i];
    }
}

// ---------------- aggregation ----------------

// acc[n][f] = b[f] + (2 * dinv[n]^2) * T[n][f]   (self-loop contribution + bias)
__global__ void k_init_acc(const float* __restrict__ T, const float* __restrict__ dinv,
                           const float* __restrict__ b, float* __restrict__ acc, int N) {
    int t = blockIdx.x * blockDim.x + threadIdx.x;
    if (t >= N * HDIM) return;
    int n = t >> 7, f = t & 127;
    float dn = dinv[n];
    acc[t] = b[f] + 2.0f * dn * dn * T[t];
}

// one wave per edge; lane handles 4 consecutive features
__global__ void k_scatter(const float* __restrict__ T, const int* __restrict__ src,
                          const int* __restrict__ dst, const float* __restrict__ norm,
                          float* __restrict__ acc, int E) {
    int t = blockIdx.x * blockDim.x + threadIdx.x;
    int e = t >> 5;
    if (e >= E) return;
    int lane = t & 31;
    int s = src[e], d = dst[e];
    float w = norm[e];
    float4 v = ((const float4*)(T + (size_t)s * HDIM))[lane];
    float* outp = acc + (size_t)d * HDIM + lane * 4;
    atomicAdd(outp + 0, w * v.x);
    atomicAdd(outp + 1, w * v.y);
    atomicAdd(outp + 2, w * v.z);
    atomicAdd(outp + 3, w * v.w);
}

__global__ void k_tanh(float* __restrict__ h, int count) {
    int t = blockIdx.x * blockDim.x + threadIdx.x;
    if (t < count) h[t] = tanhf(h[t]);
}

// ---------------- pooling ----------------

__device__ inline void atomicMaxF(float* addr, float val) {
    int* ai = (int*)addr;
    int old = __float_as_int(*addr);
    while (__int_as_float(old) < val) {
        int prev = atomicCAS(ai, old, __float_as_int(val));
        if (prev == old) break;
        old = prev;
    }
}

__global__ void k_pool_init(float* gmax, float* gsum, float* cnt, int B) {
    int t = blockIdx.x * blockDim.x + threadIdx.x;
    if (t < B * HDIM) { gmax[t] = -3.0e38f; gsum[t] = 0.0f; }
    if (t < B) cnt[t] = 0.0f;
}

__global__ void k_pool_scatter(const float* __restrict__ h, const int* __restrict__ batch,
                               float* gmax, float* gsum, float* cnt, int N) {
    int t = blockIdx.x * blockDim.x + threadIdx.x;
    if (t >= N * HDIM) return;
    int n = t >> 7, f = t & 127;
    int g = batch[n];
    float v = h[t];
    atomicMaxF(&gmax[g * HDIM + f], v);
    atomicAdd(&gsum[g * HDIM + f], v);
    if (f == 0) atomicAdd(&cnt[g], 1.0f);
}

// hidden[g] = [gmax (guarded), gmean] -> written into d_out + B
__global__ void k_pool_finalize(const float* __restrict__ gmax, const float* __restrict__ gsum,
                                const float* __restrict__ cnt, float* __restrict__ hidden, int B) {
    int t = blockIdx.x * blockDim.x + threadIdx.x;
    if (t >= B * HDIM) return;
    int g = t >> 7, f = t & 127;
    float mx = gmax[t];
    if (!(mx > -1.0e38f)) mx = 0.0f;  // empty-graph guard (mirrors isfinite check)
    float c = cnt[g];
    float mean = gsum[t] / fmaxf(c, 1.0f);
    hidden[g * (2 * HDIM) + f]        = mx;
    hidden[g * (2 * HDIM) + HDIM + f] = mean;
}

// out[g] = dot(hidden[g], Wout) + bout ; one wave per graph
__global__ void k_final_out(const float* __restrict__ hidden, const float* __restrict__ Wout,
                            const float* __restrict__ bout, float* __restrict__ out, int B) {
    int g = blockIdx.x;
    int lane = threadIdx.x;  // 32 threads
    float s = 0.0f;
    for (int f = lane; f < 2 * HDIM; f += 32) s += hidden[g * (2 * HDIM) + f] * Wout[f];
    for (int off = 16; off > 0; off >>= 1) s += __shfl_down(s, off, 32);
    if (lane == 0) out[g] = s + bout[0];
}

// ---------------- host launch ----------------

extern "C" void kernel_launch(void* const* d_in, const int* in_sizes, int n_in,
                              void* d_out, int out_size, void* d_ws, size_t ws_size,
                              hipStream_t stream) {
    const float* x    = (const float*)d_in[0];
    const int*   ei   = (const int*)d_in[1];
    const int*   bat  = (const int*)d_in[2];
    const float* ew   = (const float*)d_in[3];
    const float* Wl[4] = { (const float*)d_in[4], (const float*)d_in[6],
                           (const float*)d_in[8], (const float*)d_in[10] };
    const float* bl[4] = { (const float*)d_in[5], (const float*)d_in[7],
                           (const float*)d_in[9], (const float*)d_in[11] };
    const float* Wout = (const float*)d_in[12];
    const float* bout = (const float*)d_in[13];

    const int N = in_sizes[2];          // batch_index length = #nodes
    const int E = in_sizes[3];          // edge_weight length = #edges
    const int B = out_size / (1 + 2 * HDIM);  // 64

    const int* src = ei;
    const int* dst = ei + E;

    // workspace carve-up (16B-aligned slices)
    size_t Npad = ((size_t)N + 3) & ~(size_t)3;
    size_t Epad = ((size_t)E + 3) & ~(size_t)3;
    float* ws   = (float*)d_ws;
    float* dinv = ws;                         // Npad
    float* norm = dinv + Npad;                // Epad
    float* tbuf = norm + Epad;                // N*128
    float* bufA = tbuf + Npad * HDIM;         // N*128
    float* bufB = bufA + Npad * HDIM;         // N*128
    float* gmax = bufB + Npad * HDIM;         // B*128
    float* gsum = gmax + (size_t)B * HDIM;    // B*128
    float* cnt  = gsum + (size_t)B * HDIM;    // B

    float* out_ptr    = (float*)d_out;        // [B]
    float* hidden_ptr = out_ptr + B;          // [B, 256]

    const int T256 = 256;
    int gN   = (N + T256 - 1) / T256;
    int gE   = (E + T256 - 1) / T256;
    int gNF  = (N * HDIM + T256 - 1) / T256;
    int gEw  = ((E * 32) + T256 - 1) / T256;  // wave-per-edge
    int gBF  = (B * HDIM + T256 - 1) / T256;
    int gTile = (N + 15) / 16;

    // normalization
    k_deg_init<<<gN, T256, 0, stream>>>(dinv, N);
    k_deg_edges<<<gE, T256, 0, stream>>>(dst, ew, dinv, E);
    k_deg_finalize<<<gN, T256, 0, stream>>>(dinv, N);
    k_norm<<<gE, T256, 0, stream>>>(src, dst, ew, dinv, norm, E);

    // 4 GCN layers, ping-pong bufA/bufB
    const float* hin = x;
    float* houts[4] = { bufA, bufB, bufA, bufB };
    for (int l = 0; l < 4; ++l) {
        float* hout = houts[l];
        k_gemm_wmma<<<gTile, T256, 0, stream>>>(hin, Wl[l], tbuf, N);
        k_init_acc<<<gNF, T256, 0, stream>>>(tbuf, dinv, bl[l], hout, N);
        k_scatter<<<gEw, T256, 0, stream>>>(tbuf, src, dst, norm, hout, E);
        k_tanh<<<gNF, T256, 0, stream>>>(hout, N * HDIM);
        hin = hout;
    }

    // pooling + readout
    k_pool_init<<<gBF, T256, 0, stream>>>(gmax, gsum, cnt, B);
    k_pool_scatter<<<gNF, T256, 0, stream>>>(hin, bat, gmax, gsum, cnt, N);
    k_pool_finalize<<<gBF, T256, 0, stream>>>(gmax, gsum, cnt, hidden_ptr, B);
    k_final_out<<<B, 32, 0, stream>>>(hidden_ptr, Wout, bout, out_ptr, B);
}